// RandomShiftsAug_59115929862159
// MI455X (gfx1250) — compile-verified
//
#include <hip/hip_runtime.h>

#define PAD    4
#define CCH    9
#define OUT_H  84
#define WID    420
#define CHUNKS 105   // WID / 4

// exact parameter type from hipcc diagnostic: 'vector_size(4*sizeof(int)) int __device__ *'
typedef int v4i __attribute__((vector_size(16)));
typedef __attribute__((address_space(1))) v4i gv4i;   // global (AS1)
typedef __attribute__((address_space(3))) v4i lv4i;   // LDS    (AS3)

#if defined(__gfx1250__) && __has_builtin(__builtin_amdgcn_global_load_async_to_lds_b128)
#define USE_ASYNC_LDS 1
#else
#define USE_ASYNC_LDS 0
#endif

__global__ __launch_bounds__(256) void RandomShiftsAug_59115929862159_kernel(
    const float* __restrict__ x,      // (128, 9, 84, 420) f32
    const int*   __restrict__ shift,  // (128, 2) i32 : [:,0]=kx, [:,1]=ky
    float*       __restrict__ out,    // (128, 9, 84, 420) f32
    int rows_total)                   // 128*9*84
{
    // two source rows staged per block (one per 128-thread half = 4 waves)
    __shared__ __align__(16) float row_buf[2][WID];

    const int half = threadIdx.x >> 7;   // 0 or 1
    const int t    = threadIdx.x & 127;  // lane within half

    int row_id = blockIdx.x * 2 + half;
    if (row_id >= rows_total) row_id = rows_total - 1;  // keep barriers uniform

    const int y   = row_id % OUT_H;
    const int img = row_id / OUT_H;      // n*9 + c
    const int n   = img / CCH;

    const int kx = shift[2 * n + 0];
    const int ky = shift[2 * n + 1];

    // row map: edge-pad top (clamp to 0) then 5x H-tile wrap (tr <= 87)
    int tr = y + ky - PAD;
    tr = (tr < 0) ? 0 : tr;
    tr = (tr >= OUT_H) ? (tr - OUT_H) : tr;

    const float* src = x + (size_t)(img * OUT_H + tr) * WID;  // 16B-aligned: 420*4 % 16 == 0

    // ---- stage source row into LDS (aligned b128 async DMA chunks) ----
#if USE_ASYNC_LDS
    if (t < CHUNKS) {
        __builtin_amdgcn_global_load_async_to_lds_b128(
            (gv4i*)(src + 4 * t),
            (lv4i*)(&row_buf[half][4 * t]),
            /*offset=*/0, /*cpol=*/0);
    }
  #if __has_builtin(__builtin_amdgcn_s_wait_asynccnt)
    __builtin_amdgcn_s_wait_asynccnt(0);
  #else
    asm volatile("s_wait_asynccnt 0" ::: "memory");
  #endif
#else
    if (t < CHUNKS) {
        const float4 v = *(const float4*)(src + 4 * t);
        *(float4*)(&row_buf[half][4 * t]) = v;
    }
#endif
    __syncthreads();

    // ---- shifted, edge-clamped read from LDS; aligned b128 store ----
    if (t < CHUNKS) {
        const int b = 4 * t + kx - PAD;
        int c0 = b + 0; c0 = (c0 < 0) ? 0 : ((c0 > WID - 1) ? WID - 1 : c0);
        int c1 = b + 1; c1 = (c1 < 0) ? 0 : ((c1 > WID - 1) ? WID - 1 : c1);
        int c2 = b + 2; c2 = (c2 < 0) ? 0 : ((c2 > WID - 1) ? WID - 1 : c2);
        int c3 = b + 3; c3 = (c3 < 0) ? 0 : ((c3 > WID - 1) ? WID - 1 : c3);

        float4 o;
        o.x = row_buf[half][c0];
        o.y = row_buf[half][c1];
        o.z = row_buf[half][c2];
        o.w = row_buf[half][c3];

        *(float4*)(out + (size_t)row_id * WID + 4 * t) = o;
    }
}

extern "C" void kernel_launch(void* const* d_in, const int* in_sizes, int n_in,
                              void* d_out, int out_size, void* d_ws, size_t ws_size,
                              hipStream_t stream) {
    const float* x     = (const float*)d_in[0];
    const int*   shift = (const int*)d_in[1];
    float*       out   = (float*)d_out;

    const int n_batch    = in_sizes[1] / 2;          // 128
    const int rows_total = n_batch * CCH * OUT_H;    // 96768
    const int n_blocks   = (rows_total + 1) / 2;     // 2 rows per 256-thread block

    RandomShiftsAug_59115929862159_kernel<<<dim3(n_blocks), dim3(256), 0, stream>>>(
        x, shift, out, rows_total);
}